// GNN_76003741270414
// MI455X (gfx1250) — compile-verified
//
#include <hip/hip_runtime.h>

#define NEG_SLOPE 0.2f
#define HEADS 2
#define CPH   16
#define HC    32
#define HID   32
#define OUTC  2

typedef float v2f __attribute__((ext_vector_type(2)));
typedef float v8f __attribute__((ext_vector_type(8)));

// Order-preserving float -> uint key (for atomicMax-based segment_max)
__device__ __forceinline__ unsigned fkey(float f) {
  unsigned u = __float_as_uint(f);
  return (u & 0x80000000u) ? ~u : (u | 0x80000000u);
}
__device__ __forceinline__ float funkey(unsigned k) {
  unsigned u = (k & 0x80000000u) ? (k & 0x7FFFFFFFu) : ~k;
  return __uint_as_float(u);
}

// ---------------- K1: per-node linear transforms xl = xW_l + b_l, xr = xW_r + b_r
__global__ void node_transform(const float* __restrict__ x,
                               const float* __restrict__ Wl, const float* __restrict__ bl,
                               const float* __restrict__ Wr, const float* __restrict__ br,
                               float* __restrict__ xl, float* __restrict__ xr, int n) {
  int idx = blockIdx.x * blockDim.x + threadIdx.x;
  if (idx >= n * HC) return;
  int i = idx >> 5, j = idx & 31;
  float x0 = x[i * 2 + 0], x1 = x[i * 2 + 1];
  xl[idx] = fmaf(x0, Wl[j], fmaf(x1, Wl[HC + j], bl[j]));
  xr[idx] = fmaf(x0, Wr[j], fmaf(x1, Wr[HC + j], br[j]));
}

// ---------------- K2: per-edge logits + scatter-max over dst
__global__ void edge_logits(const int* __restrict__ ei, const float* __restrict__ xl,
                            const float* __restrict__ xr, const float* __restrict__ att,
                            float* __restrict__ logits, unsigned* __restrict__ mkey,
                            int E_, int Et) {
  int e = blockIdx.x * blockDim.x + threadIdx.x;
  if (e >= Et) return;
  int s, d;
  if (e < E_) { s = ei[e]; d = ei[E_ + e]; } else { s = d = e - E_; }
  float a[HC], b[HC];
  const float4* pa = (const float4*)(xl + (size_t)s * HC);
  const float4* pb = (const float4*)(xr + (size_t)d * HC);
#pragma unroll
  for (int i = 0; i < HC / 4; ++i) {
    float4 va = pa[i], vb = pb[i];
    a[4*i+0]=va.x; a[4*i+1]=va.y; a[4*i+2]=va.z; a[4*i+3]=va.w;
    b[4*i+0]=vb.x; b[4*i+1]=vb.y; b[4*i+2]=vb.z; b[4*i+3]=vb.w;
  }
#pragma unroll
  for (int h = 0; h < HEADS; ++h) {
    float acc = 0.f;
#pragma unroll
    for (int c = 0; c < CPH; ++c) {
      float v = a[h*CPH + c] + b[h*CPH + c];
      v = (v > 0.f) ? v : NEG_SLOPE * v;            // leaky_relu
      acc = fmaf(att[h*CPH + c], v, acc);
    }
    logits[(size_t)e * HEADS + h] = acc;
    atomicMax(&mkey[d * HEADS + h], fkey(acc));
  }
}

// ---------------- K3: exp(logit - max) + scatter-sum denominator
__global__ void edge_softmax_num(const int* __restrict__ ei,
                                 float* __restrict__ logits,          // in: logit, out: exp
                                 const unsigned* __restrict__ mkey,
                                 float* __restrict__ denom, int E_, int Et) {
  int idx = blockIdx.x * blockDim.x + threadIdx.x;
  if (idx >= Et * HEADS) return;
  int e = idx >> 1, h = idx & 1;
  int d = (e < E_) ? ei[E_ + e] : (e - E_);
  float m = funkey(mkey[d * HEADS + h]);
  if (!(fabsf(m) < 3.0e38f)) m = 0.f;               // isfinite guard (NaN compares false)
  float ex = expf(logits[idx] - m);
  logits[idx] = ex;
  atomicAdd(&denom[d * HEADS + h], ex);
}

// ---------------- K4: alpha-weighted message scatter
__global__ void edge_scatter(const int* __restrict__ ei, const float* __restrict__ xl,
                             const float* __restrict__ expl, const float* __restrict__ denom,
                             float* __restrict__ feat, int E_, int Et) {
  int idx = blockIdx.x * blockDim.x + threadIdx.x;
  if (idx >= Et * HEADS) return;
  int e = idx >> 1, h = idx & 1;
  int s, d;
  if (e < E_) { s = ei[e]; d = ei[E_ + e]; } else { s = d = e - E_; }
  float alpha = expl[idx] / (denom[d * HEADS + h] + 1e-16f);
  const float* xs = xl + (size_t)s * HC + h * CPH;
  float* fd = feat + (size_t)d * HC + h * CPH;
#pragma unroll
  for (int c = 0; c < CPH; ++c)
    atomicAdd(&fd[c], xs[c] * alpha);
}

// ---------------- K5: MLP decoder with f32 WMMA (16x16x4)
// One wave computes 16 rows: (feat+bias)[16x32] @ W1[32x32] -> relu -> @ W2[32x2]
__global__ __launch_bounds__(256) void mlp_wmma(
    const float* __restrict__ feat, const float* __restrict__ bias,
    const float* __restrict__ W1, const float* __restrict__ b1v,
    const float* __restrict__ W2, const float* __restrict__ b2v,
    float* __restrict__ out, int n) {
  __shared__ float hbuf[8][16][HID];
  int lane = threadIdx.x & 31;
  int wave = threadIdx.x >> 5;
  int rowBase = (blockIdx.x * 8 + wave) * 16;

  int am  = lane & 15;        // A tile row / B tile col
  int akb = (lane >> 4) * 2;  // K sub-offset within K=4 chunk (0 or 2)

  v8f c0 = {};  // output cols 0..15
  v8f c1 = {};  // output cols 16..31
#pragma unroll
  for (int k0 = 0; k0 < HC; k0 += 4) {
    int row = rowBase + am; if (row >= n) row = n - 1;     // clamp: keep EXEC all-1s
    const float* ap = feat + (size_t)row * HC + k0 + akb;
    v2f a; a.x = ap[0] + bias[k0 + akb];
           a.y = ap[1] + bias[k0 + akb + 1];
    v2f bA, bB;
    bA.x = W1[(k0 + akb)     * HID + am];
    bA.y = W1[(k0 + akb + 1) * HID + am];
    bB.x = W1[(k0 + akb)     * HID + 16 + am];
    bB.y = W1[(k0 + akb + 1) * HID + 16 + am];
    c0 = __builtin_amdgcn_wmma_f32_16x16x4_f32(false, a, false, bA, (short)0, c0, false, false);
    c1 = __builtin_amdgcn_wmma_f32_16x16x4_f32(false, a, false, bB, (short)0, c1, false, false);
  }
  // D layout: VGPR v, lanes 0-15 -> M=v, N=lane; lanes 16-31 -> M=8+v, N=lane-16
  int drow = (lane >> 4) * 8;
  int dcol = lane & 15;
#pragma unroll
  for (int v = 0; v < 8; ++v) {
    hbuf[wave][drow + v][dcol]      = fmaxf(c0[v] + b1v[dcol],      0.f);
    hbuf[wave][drow + v][16 + dcol] = fmaxf(c1[v] + b1v[16 + dcol], 0.f);
  }
  __syncthreads();
  // 32 lanes -> (row m, out-channel o); tiny 32->2 projection in VALU
  int m = lane & 15, o = lane >> 4;
  float acc = b2v[o];
#pragma unroll
  for (int k = 0; k < HID; ++k)
    acc = fmaf(hbuf[wave][m][k], W2[k * OUTC + o], acc);
  int row = rowBase + m;
  if (row < n) out[(size_t)row * OUTC + o] = acc;
}

extern "C" void kernel_launch(void* const* d_in, const int* in_sizes, int n_in,
                              void* d_out, int out_size, void* d_ws, size_t ws_size,
                              hipStream_t stream) {
  const float* x    = (const float*)d_in[0];
  const int*   ei   = (const int*)  d_in[1];
  const float* Wl   = (const float*)d_in[2];
  const float* bl   = (const float*)d_in[3];
  const float* Wr   = (const float*)d_in[4];
  const float* br   = (const float*)d_in[5];
  const float* att  = (const float*)d_in[6];
  const float* bias = (const float*)d_in[7];
  const float* W1   = (const float*)d_in[8];
  const float* b1   = (const float*)d_in[9];
  const float* W2   = (const float*)d_in[10];
  const float* b2   = (const float*)d_in[11];
  float* out = (float*)d_out;

  int n  = in_sizes[0] / 2;   // NODE_DIM = 2
  int E_ = in_sizes[1] / 2;   // edge_index is [2, E]
  int Et = E_ + n;            // + self loops

  size_t N = (size_t)n;
  float*    xl     = (float*)d_ws;
  float*    xr     = xl + N * HC;
  unsigned* mkey   = (unsigned*)(xr + N * HC);
  float*    denom  = (float*)(mkey + N * HEADS);
  float*    feat   = denom + N * HEADS;
  float*    logits = feat + N * HC;

  // zero atomic accumulators (mkey | denom | feat are contiguous)
  hipMemsetAsync(mkey, 0, (N * HEADS + N * HEADS + N * HC) * sizeof(float), stream);

  const int thr = 256;
  node_transform  <<<(n * HC + thr - 1) / thr,        thr, 0, stream>>>(x, Wl, bl, Wr, br, xl, xr, n);
  edge_logits     <<<(Et + thr - 1) / thr,            thr, 0, stream>>>(ei, xl, xr, att, logits, mkey, E_, Et);
  edge_softmax_num<<<(Et * HEADS + thr - 1) / thr,    thr, 0, stream>>>(ei, logits, mkey, denom, E_, Et);
  edge_scatter    <<<(Et * HEADS + thr - 1) / thr,    thr, 0, stream>>>(ei, xl, logits, denom, feat, E_, Et);

  int nwaves = (n + 15) / 16;
  int blocks = (nwaves + 7) / 8;
  mlp_wmma<<<blocks, 256, 0, stream>>>(feat, bias, W1, b1, W2, b2, out, n);
}